// GaussianImage_Cholesky_12249246728856
// MI455X (gfx1250) — compile-verified
//
#include <hip/hip_runtime.h>

#define IMG_H 512
#define IMG_W 512
#define NG    10240
#define MAXG  2048   // per-tile list capacity (expected ~33, Poisson; 2048 is untouchable)

typedef float v2f __attribute__((ext_vector_type(2)));
typedef float v8f __attribute__((ext_vector_type(8)));

// ---------------------------------------------------------------------------
// Kernel 1: per-gaussian precompute.
// params[g] = {cx, cy, 0.5*A, B, 0.5*C, op, pad, pad}  (8 floats, 32B aligned)
// bbox[g]   = {cx, cy, ex, ey}   (cull-only data: 16B instead of 32B per test)
// col4[g]   = {r, g, b, 0}
// ex/ey are conservative half-extents of the alpha>=1/255 footprint (+1 px slop);
// set to -1e30 when the gaussian can never pass the threshold (op*255 <= 1).
// ---------------------------------------------------------------------------
__global__ __launch_bounds__(256) void gi_prep(const float* __restrict__ xyz,
                                               const float* __restrict__ chol,
                                               const float* __restrict__ opac,
                                               const float* __restrict__ feat,
                                               float* __restrict__ P,
                                               float* __restrict__ BB,
                                               float* __restrict__ C4) {
  int g = blockIdx.x * 256 + threadIdx.x;
  if (g >= NG) return;
  float mx = tanhf(xyz[2 * g + 0]);
  float my = tanhf(xyz[2 * g + 1]);
  float cx = 0.5f * IMG_W * (mx + 1.0f);
  float cy = 0.5f * IMG_H * (my + 1.0f);
  float l1 = chol[3 * g + 0] + 0.5f;
  float l2 = chol[3 * g + 1];
  float l3 = chol[3 * g + 2] + 0.5f;
  float c00 = l1 * l1;
  float c01 = l1 * l2;
  float c11 = l2 * l2 + l3 * l3;
  float det = c00 * c11 - c01 * c01;
  float inv = 1.0f / det;
  float A = c11 * inv;
  float Bc = -c01 * inv;
  float Cc = c00 * inv;
  float op = opac[g];
  // alpha >= 1/255  <=>  sigma <= T = ln(255*op)
  float T = __logf(255.0f * op);
  float ex, ey;
  if (T > 0.0f && det > 0.0f) {
    // max |dx| over {sigma<=T} = sqrt(2*T*Sigma_xx) = sqrt(2*T*c00); same for y with c11
    ex = sqrtf(2.0f * T * c00) + 1.0f;
    ey = sqrtf(2.0f * T * c11) + 1.0f;
  } else {
    ex = -1e30f;
    ey = -1e30f;
  }
  float4* p4 = (float4*)(P + 8 * g);
  p4[0] = make_float4(cx, cy, 0.5f * A, Bc);
  p4[1] = make_float4(0.5f * Cc, op, 0.0f, 0.0f);
  ((float4*)BB)[g] = make_float4(cx, cy, ex, ey);
  ((float4*)C4)[g] = make_float4(feat[3 * g + 0], feat[3 * g + 1], feat[3 * g + 2], 0.0f);
}

// ---------------------------------------------------------------------------
// alpha evaluation for one gaussian at one pixel (branch-free: v_cndmask, so
// EXEC stays full for the WMMAs).
// ---------------------------------------------------------------------------
__device__ __forceinline__ float gi_alpha(float4 pa, float4 pb, float px, float py) {
  float dx = pa.x - px;
  float dy = pa.y - py;
  float s = pa.z * dx * dx;          // 0.5*A*dx^2
  s = fmaf(pb.x, dy * dy, s);        // + 0.5*C*dy^2
  s = fmaf(pa.w, dx * dy, s);        // + B*dx*dy
  float al = pb.y * __expf(-s);      // op * exp(-sigma)  -> v_exp_f32 (trans pipe)
  al = fminf(al, 0.999f);
  bool ok = (s >= 0.0f) && (al >= (1.0f / 255.0f));
  return ok ? al : 0.0f;
}

// ---------------------------------------------------------------------------
// Kernel 2: one workgroup (256 thr = 8 waves) per 16x16 pixel tile.
// Phase A: deterministic ordered cull of all N gaussians into an LDS list,
//          software-pipelined (next chunk's bboxes load while this chunk's
//          ballot/prefix/compaction + barriers run).
// Phase B: wave w owns image rows 2w and 2w+1 (16 px each = WMMA M dim).
//          K-loop in steps of 4 gaussians; A = alphas (16px x 4), B = colors
//          (4 x 16ch, cols>=3 zero), C/D = f32 running tile, accumulated with
//          v_wmma_f32_16x16x4_f32 on the matrix pipe.
// ---------------------------------------------------------------------------
__global__ __launch_bounds__(256) void gi_render(const float* __restrict__ P,
                                                 const float* __restrict__ BB,
                                                 const float* __restrict__ C4,
                                                 float* __restrict__ out) {
  __shared__ int s_list[MAXG + 4];
  __shared__ int s_cnt[8];
  __shared__ int s_total;

  const int tid = threadIdx.x;
  const int lane = tid & 31;
  const int wv = tid >> 5;
  const int tx0 = blockIdx.x * 16;
  const int ty0 = blockIdx.y * 16;
  const float tcx = (float)tx0 + 8.0f;
  const float tcy = (float)ty0 + 8.0f;

  if (tid == 0) s_total = 0;
  __syncthreads();

  // -------- Phase A: ordered cull (deterministic compaction), pipelined --------
  float4 bb = ((const float4*)BB)[tid];  // chunk 0
  for (int base = 0; base < NG; base += 256) {
    float4 bbn = bb;
    if (base + 256 < NG) bbn = ((const float4*)BB)[base + 256 + tid];  // prefetch next
    bool vis = (fabsf(bb.x - tcx) <= bb.z + 8.0f) && (fabsf(bb.y - tcy) <= bb.w + 8.0f);
    unsigned m = __builtin_amdgcn_ballot_w32(vis);
    if (lane == 0) s_cnt[wv] = __popc(m);
    __syncthreads();
    int tot0 = s_total;
    int pre = tot0, all = 0;
    for (int w = 0; w < 8; ++w) {
      int c = s_cnt[w];
      if (w < wv) pre += c;
      all += c;
    }
    if (vis) {
      int idx = pre + __popc(m & ((1u << lane) - 1u));
      if (idx < MAXG) s_list[idx] = base + tid;
    }
    __syncthreads();
    if (tid == 0) s_total = (tot0 + all < MAXG) ? (tot0 + all) : MAXG;
    bb = bbn;
  }
  __syncthreads();
  const int total = s_total;
  if (tid < 4) s_list[total + tid] = 0;  // pad so k+3 reads are always valid
  __syncthreads();

  // -------- Phase B: WMMA accumulate --------
  const int n = lane & 15;   // WMMA N (channel) for B/D, and M (pixel column) for A
  const int h = lane >> 4;   // half-wave supplies K = 2h, 2h+1
  const int ci = (n < 3) ? n : 3;             // col4[g][3] == 0 -> zero B columns 3..15
  const float px = (float)(tx0 + n) + 0.5f;
  const float py0 = (float)(ty0 + 2 * wv) + 0.5f;
  const float py1 = py0 + 1.0f;

  v8f acc0 = {0, 0, 0, 0, 0, 0, 0, 0};
  v8f acc1 = {0, 0, 0, 0, 0, 0, 0, 0};

  for (int k = 0; k < total; k += 4) {
    int i0 = s_list[k + 2 * h];              // gaussian for K = 2h   (lanes 0-15: k, k+1)
    int i1 = s_list[k + 2 * h + 1];          // gaussian for K = 2h+1 (lanes16-31: k+2,k+3)
    float4 p0a = *(const float4*)(P + 8 * i0);     // broadcast loads (2 addrs per wave)
    float4 p0b = *(const float4*)(P + 8 * i0 + 4);
    float4 p1a = *(const float4*)(P + 8 * i1);
    float4 p1b = *(const float4*)(P + 8 * i1 + 4);
    float m0 = (k + 2 * h < total) ? 1.0f : 0.0f;
    float m1 = (k + 2 * h + 1 < total) ? 1.0f : 0.0f;

    v2f a0, a1, b;
    a0.x = gi_alpha(p0a, p0b, px, py0) * m0;   // A VGPR0 = K(2h)
    a0.y = gi_alpha(p1a, p1b, px, py0) * m1;   // A VGPR1 = K(2h+1)
    a1.x = gi_alpha(p0a, p0b, px, py1) * m0;
    a1.y = gi_alpha(p1a, p1b, px, py1) * m1;
    b.x = C4[4 * i0 + ci];                     // B VGPR0 = col[K(2h)][n]
    b.y = C4[4 * i1 + ci];                     // B VGPR1 = col[K(2h+1)][n]

    acc0 = __builtin_amdgcn_wmma_f32_16x16x4_f32(false, a0, false, b, (short)0, acc0,
                                                 false, false);
    acc1 = __builtin_amdgcn_wmma_f32_16x16x4_f32(false, a1, false, b, (short)0, acc1,
                                                 false, false);
  }

  // -------- Phase C: writeout. D VGPR r holds D[M=r+8h][N=n]; out is (1,3,H,W). --------
  if (n < 3) {
    float* o = out + n * (IMG_H * IMG_W) + (ty0 + 2 * wv) * IMG_W + tx0 + 8 * h;
#define CL(v) fminf(fmaxf((v), 0.0f), 1.0f)
    *(float4*)(o + 0) = make_float4(CL(acc0[0]), CL(acc0[1]), CL(acc0[2]), CL(acc0[3]));
    *(float4*)(o + 4) = make_float4(CL(acc0[4]), CL(acc0[5]), CL(acc0[6]), CL(acc0[7]));
    o += IMG_W;
    *(float4*)(o + 0) = make_float4(CL(acc1[0]), CL(acc1[1]), CL(acc1[2]), CL(acc1[3]));
    *(float4*)(o + 4) = make_float4(CL(acc1[4]), CL(acc1[5]), CL(acc1[6]), CL(acc1[7]));
#undef CL
  }
}

// ---------------------------------------------------------------------------
extern "C" void kernel_launch(void* const* d_in, const int* in_sizes, int n_in,
                              void* d_out, int out_size, void* d_ws, size_t ws_size,
                              hipStream_t stream) {
  const float* xyz = (const float*)d_in[0];
  const float* chol = (const float*)d_in[1];
  const float* opac = (const float*)d_in[2];
  const float* feat = (const float*)d_in[3];
  float* params = (float*)d_ws;                 // NG*8 floats = 320 KB
  float* bbox = params + (size_t)NG * 8;        // NG*4 floats = 160 KB
  float* col4 = bbox + (size_t)NG * 4;          // NG*4 floats = 160 KB
  float* out = (float*)d_out;

  gi_prep<<<NG / 256, 256, 0, stream>>>(xyz, chol, opac, feat, params, bbox, col4);
  dim3 grid(IMG_W / 16, IMG_H / 16);
  gi_render<<<grid, 256, 0, stream>>>(params, bbox, col4, out);
}